// ModulePropagationModel_78769700208709
// MI455X (gfx1250) — compile-verified
//
#include <hip/hip_runtime.h>

typedef __attribute__((ext_vector_type(2))) float v2f;
typedef __attribute__((ext_vector_type(8))) float v8f;

namespace {
constexpr int B_   = 16;
constexpr int G_   = 5000;
constexpr int K_   = 256;
constexpr int HID_ = 64;
constexpr int STEPS_ = 10;
constexpr float ALPHA_ = 0.1f;
constexpr int NTILES_ = (G_ + 15) / 16;            // 313 gene tiles of 16
constexpr int WPB_ = 4;                            // waves per block (kernel 3)
constexpr int NBLK_ = (NTILES_ + WPB_ - 1) / WPB_; // 79
}

// ---------------------------------------------------------------------------
// Kernel 1: t_mod[b,k] = max_g t[b,g] * M[g,k]
// One block per batch row b; lane == module k; M row reads fully coalesced.
// ---------------------------------------------------------------------------
__global__ __launch_bounds__(256) void pool_kernel(const float* __restrict__ t,
                                                   const float* __restrict__ M,
                                                   float* __restrict__ t_mod) {
  const int b = blockIdx.x;
  const int k = threadIdx.x;
  const float* __restrict__ trow = t + (size_t)b * G_;
  float acc = -3.0e38f;
#pragma unroll 4
  for (int g = 0; g < G_; ++g) {
    acc = fmaxf(acc, trow[g] * M[(size_t)g * K_ + k]);
  }
  t_mod[b * K_ + k] = acc;
}

// ---------------------------------------------------------------------------
// Kernel 2: APPNP, Z resident in LDS. 16 waves; wave w owns output columns
// [16w, 16w+16) of Z. Each step: Z_new = 0.9*(Z@A) + 0.1*t_mod via
// v_wmma_f32_16x16x4_f32 (64 k-steps). Also emits ce_dot[b] and Z to global.
// ---------------------------------------------------------------------------
__global__ __launch_bounds__(512) void appnp_kernel(const float* __restrict__ Amat,
                                                    const float* __restrict__ t_mod,
                                                    const int*   __restrict__ cell_idx,
                                                    const float* __restrict__ cell_emb,
                                                    const float* __restrict__ W2,
                                                    float* __restrict__ Zout,
                                                    float* __restrict__ ce_dot) {
  __shared__ float Zs[B_ * K_];  // 16 KB

  const int tid  = threadIdx.x;
  const int lane = tid & 31;
  const int wave = tid >> 5;          // 0..15
  const int n0   = wave * 16;
  const int col  = n0 + (lane & 15);  // this lane's C-layout column
  const int rowoff = (lane >> 4) * 8; // C layout: upper half-wave holds rows 8..15
  const int koff = (lane >> 4) * 2;   // A/B layout: upper half-wave holds K+2,K+3
  const int arow = lane & 15;         // A-operand row (batch index)

  // Init Z = t_mod
  for (int i = tid; i < B_ * K_; i += 512) Zs[i] = t_mod[i];
  __syncthreads();

  // Preload this lane's t_mod tile in C layout (constant across steps)
  float tm[8];
#pragma unroll
  for (int r = 0; r < 8; ++r) tm[r] = t_mod[(r + rowoff) * K_ + col];

  for (int step = 0; step < STEPS_; ++step) {
    v8f acc = {};
#pragma unroll 4
    for (int k0 = 0; k0 < K_; k0 += 4) {
      // A operand: Z rows (from LDS), contiguous K pair -> b64 ds load
      v2f a = *(const v2f*)&Zs[arow * K_ + k0 + koff];
      // B operand: A matrix, rows k0+koff / k0+koff+1 at this column
      v2f b;
      b.x = Amat[(size_t)(k0 + koff) * K_ + col];
      b.y = Amat[(size_t)(k0 + koff + 1) * K_ + col];
      acc = __builtin_amdgcn_wmma_f32_16x16x4_f32(
          /*neg_a=*/false, a, /*neg_b=*/false, b,
          /*c_mod=*/(short)0, acc, /*reuse_a=*/false, /*reuse_b=*/false);
    }
    __syncthreads();  // everyone done reading old Z
#pragma unroll
    for (int r = 0; r < 8; ++r)
      Zs[(r + rowoff) * K_ + col] = (1.0f - ALPHA_) * acc[r] + ALPHA_ * tm[r];
    __syncthreads();  // new Z visible
  }

  // Publish Z for kernel 3
  for (int i = tid; i < B_ * K_; i += 512) Zout[i] = Zs[i];

  // ce_dot[b] = cell_emb[cell_idx[b]] . W2  (second layer contribution of the
  // post-ReLU embedding add is per-batch constant)
  if (tid < B_) {
    const int c = cell_idx[tid];
    float s = 0.0f;
#pragma unroll 8
    for (int h = 0; h < HID_; ++h) s += cell_emb[c * HID_ + h] * W2[h];
    ce_dot[tid] = s;
  }
}

// ---------------------------------------------------------------------------
// Kernel 3: z_gene tile = Z @ M^T via WMMA (B operand pairs are contiguous in
// M rows -> global_load_b64), then MLP fused on the accumulator registers:
// y = sum_h relu(ctl*W1[0,h] + t*W1[1,h] + z*W1[2,h] + b1[h]) * W2[h]
//     + ce_dot[b] + b2.
// One wave per 16-gene tile.
// ---------------------------------------------------------------------------
__global__ __launch_bounds__(WPB_ * 32) void out_kernel(const float* __restrict__ ctl,
                                                        const float* __restrict__ t,
                                                        const float* __restrict__ M,
                                                        const float* __restrict__ Z,
                                                        const float* __restrict__ W1,
                                                        const float* __restrict__ b1,
                                                        const float* __restrict__ W2,
                                                        const float* __restrict__ b2,
                                                        const float* __restrict__ ce_dot,
                                                        float* __restrict__ y) {
  const int wave_g = blockIdx.x * WPB_ + (threadIdx.x >> 5);
  if (wave_g >= NTILES_) return;  // wave-uniform: EXEC stays all-ones

  const int lane = threadIdx.x & 31;
  const int n0   = wave_g * 16;
  const int colg = n0 + (lane & 15);
  const int gcl  = (colg < G_) ? colg : (G_ - 1);  // clamp loads; stores guarded
  const int koff = (lane >> 4) * 2;
  const int arow = lane & 15;

  v8f acc = {};
#pragma unroll 4
  for (int k0 = 0; k0 < K_; k0 += 4) {
    v2f a = *(const v2f*)(Z + arow * K_ + k0 + koff);                 // Z row pair (L2 hot)
    v2f b = *(const v2f*)(M + (size_t)gcl * K_ + k0 + koff);          // M[n][k..k+1] b64
    acc = __builtin_amdgcn_wmma_f32_16x16x4_f32(
        false, a, false, b, (short)0, acc, false, false);
  }

  // C layout: vgpr r, lane -> (b = r + rowoff, g = colg)
  const int rowoff = (lane >> 4) * 8;
  float cv[8], tv[8], s[8];
#pragma unroll
  for (int r = 0; r < 8; ++r) {
    const int b = r + rowoff;
    cv[r] = ctl[(size_t)b * G_ + gcl];
    tv[r] = t[(size_t)b * G_ + gcl];
    s[r]  = 0.0f;
  }

  const float* __restrict__ w1a = W1;            // W1[0, :]
  const float* __restrict__ w1b = W1 + HID_;     // W1[1, :]
  const float* __restrict__ w1c = W1 + 2 * HID_; // W1[2, :]
#pragma unroll 4
  for (int h = 0; h < HID_; ++h) {
    const float wa = w1a[h], wb = w1b[h], wc = w1c[h];
    const float bh = b1[h],  w2h = W2[h];
#pragma unroll
    for (int r = 0; r < 8; ++r) {
      float pre = fmaf(cv[r], wa, fmaf(tv[r], wb, fmaf(acc[r], wc, bh)));
      s[r] = fmaf(fmaxf(pre, 0.0f), w2h, s[r]);
    }
  }

  if (colg < G_) {
    const float bias = b2[0];
#pragma unroll
    for (int r = 0; r < 8; ++r) {
      const int b = r + rowoff;
      y[(size_t)b * G_ + colg] = s[r] + ce_dot[b] + bias;
    }
  }
}

// ---------------------------------------------------------------------------
// Launch. Input order: ctl, drug_targets, cell_idx, drug_fp, M, A, W1, b1,
// cell_emb, W2, b2. Output: y [16, 5000] f32.
// Workspace: t_mod (16KB) | Z (16KB) | ce_dot (64B).
// ---------------------------------------------------------------------------
extern "C" void kernel_launch(void* const* d_in, const int* in_sizes, int n_in,
                              void* d_out, int out_size, void* d_ws, size_t ws_size,
                              hipStream_t stream) {
  const float* ctl      = (const float*)d_in[0];
  const float* targets  = (const float*)d_in[1];
  const int*   cell_idx = (const int*)d_in[2];
  // d_in[3] = drug_fp (unused by the reference computation)
  const float* M        = (const float*)d_in[4];
  const float* A        = (const float*)d_in[5];
  const float* W1       = (const float*)d_in[6];
  const float* b1       = (const float*)d_in[7];
  const float* cell_emb = (const float*)d_in[8];
  const float* W2       = (const float*)d_in[9];
  const float* b2       = (const float*)d_in[10];
  float* y = (float*)d_out;

  float* t_mod  = (float*)d_ws;                       // 16*256 f32
  float* Zbuf   = t_mod + B_ * K_;                    // 16*256 f32
  float* ce_dot = Zbuf + B_ * K_;                     // 16 f32

  pool_kernel<<<B_, 256, 0, stream>>>(targets, M, t_mod);
  appnp_kernel<<<1, 512, 0, stream>>>(A, t_mod, cell_idx, cell_emb, W2, Zbuf, ce_dot);
  out_kernel<<<NBLK_, WPB_ * 32, 0, stream>>>(ctl, targets, M, Zbuf, W1, b1, W2, b2,
                                              ce_dot, y);
}